// MaxMinRelOut_25374666784829
// MI455X (gfx1250) — compile-verified
//
#include <hip/hip_runtime.h>
#include <stdint.h>

// Problem dims (fixed by the reference): B=256, N=512, M=512
#define B_DIM 256
#define N_DIM 512
#define M_DIM 512

#if defined(__gfx1250__) && __has_builtin(__builtin_amdgcn_tensor_load_to_lds)
#define USE_TDM 1
#else
#define USE_TDM 0
#endif

typedef __attribute__((ext_vector_type(4))) unsigned int v4u;
typedef __attribute__((ext_vector_type(8))) int v8i;
typedef __attribute__((ext_vector_type(4))) int v4i;

#if USE_TDM
// Issue one TDM 2D tile load: tile (tile_d0 x tile_d1) of f32 from a row-major
// tensor whose contiguous dim has length tensor_d0 and row stride stride0
// (elements). LDS destination receives the tile densely packed
// [tile_d1][tile_d0]. Descriptor bitfields per CDNA5 ISA ch.8 (D# groups).
__device__ __forceinline__ void tdm_issue_2d(void* lds_dst, const void* gsrc,
                                             uint32_t tensor_d0, uint32_t tensor_d1,
                                             uint32_t tile_d0, uint32_t tile_d1,
                                             uint32_t stride0_elems) {
  // Generic (flat) shared pointer: low 32 bits are the LDS byte offset.
  uint32_t lds_off = (uint32_t)(uintptr_t)lds_dst;
  uint64_t ga = (uint64_t)(uintptr_t)gsrc;
  v4u g0;
  g0[0] = 1u;                                             // count=1, user mode
  g0[1] = lds_off;                                        // lds_addr
  g0[2] = (uint32_t)(ga & 0xFFFFFFFFull);                 // global_addr[31:0]
  g0[3] = (uint32_t)((ga >> 32) & 0x01FFFFFFull)          // global_addr[56:32]
          | (2u << 30);                                   // type = 2 ("image")
  v8i g1;
  g1[0] = (int)(2u << 16);                                // data_size = 4 bytes
  g1[1] = (int)((tensor_d0 & 0xFFFFu) << 16);             // tensor_dim0[15:0]
  g1[2] = (int)(((tensor_d0 >> 16) & 0xFFFFu) |
                ((tensor_d1 & 0xFFFFu) << 16));           // dim0 hi | dim1 lo
  g1[3] = (int)(((tensor_d1 >> 16) & 0xFFFFu) |
                ((tile_d0 & 0xFFFFu) << 16));             // dim1 hi | tile_dim0
  g1[4] = (int)(tile_d1 & 0xFFFFu);                       // tile_dim1 (tile_dim2=0)
  g1[5] = (int)stride0_elems;                             // tensor_dim0_stride lo
  g1[6] = 0;                                              // stride hi / dim1_stride
  g1[7] = 0;
  v4i gz4 = {0, 0, 0, 0};                                 // groups 2/3 unused (2D)
  v8i gz8 = {0, 0, 0, 0, 0, 0, 0, 0};                     // extra group (clang-23 6-arg form)
  __builtin_amdgcn_tensor_load_to_lds(g0, g1, gz4, gz4, gz8, 0);
}
#else
__device__ __forceinline__ void coop_load_2d(float* lds, const float* g,
                                             int tile_d0, int tile_d1, int stride) {
  const int total = tile_d0 * tile_d1;
  for (int i = threadIdx.x; i < total; i += blockDim.x)
    lds[i] = g[(i / tile_d0) * stride + (i % tile_d0)];
}
#endif

__device__ __forceinline__ void load_tile(float* lds, const float* g,
                                          uint32_t td0, uint32_t td1,
                                          uint32_t tile0, uint32_t tile1,
                                          uint32_t stride) {
#if USE_TDM
  if (threadIdx.x < 32) tdm_issue_2d(lds, g, td0, td1, tile0, tile1, stride);
#else
  coop_load_2d(lds, g, (int)tile0, (int)tile1, (int)stride);
#endif
}

__device__ __forceinline__ void tile_fence() {
#if USE_TDM
  if (threadIdx.x < 32) __builtin_amdgcn_s_wait_tensorcnt((short)0);
#endif
  __syncthreads();
}

// ---- Kernel B: rel_x[n,m] = sum_b min(x[b,n], t[b,m]) / sum_b x[b,n] -------
// grid (M/32, N/32), block 256. LDS: x-tile 256x32 + t-tile 256x32 = 64 KB.
// The denominator (sx) is folded into the main loop: xs values are already in
// registers for the min, so the extra accumulate is nearly free VALU.
__global__ __launch_bounds__(256) void relx_kernel(const float* __restrict__ x,
                                                   const float* __restrict__ t,
                                                   float* __restrict__ relx) {
  __shared__ float xs[B_DIM * 32];   // [b][n_local]
  __shared__ float ts[B_DIM * 32];   // [b][m_local]
  const int m0 = blockIdx.x * 32;
  const int n0 = blockIdx.y * 32;
  load_tile(xs, x + n0, N_DIM, B_DIM, 32, B_DIM, N_DIM);
  load_tile(ts, t + m0, M_DIM, B_DIM, 32, B_DIM, M_DIM);
  tile_fence();

  const int ml = threadIdx.x & 31;
  const int nl = (threadIdx.x >> 5) * 4;       // 4 consecutive n rows per thread
  float a0 = 0.f, a1 = 0.f, a2 = 0.f, a3 = 0.f;
  float s0 = 0.f, s1 = 0.f, s2 = 0.f, s3 = 0.f;
  for (int b = 0; b < B_DIM; ++b) {
    const float tv = ts[b * 32 + ml];          // conflict-free across lanes
    const float x0 = xs[b * 32 + nl + 0];      // LDS broadcast across lanes
    const float x1 = xs[b * 32 + nl + 1];
    const float x2 = xs[b * 32 + nl + 2];
    const float x3 = xs[b * 32 + nl + 3];
    a0 += fminf(x0, tv); s0 += x0;
    a1 += fminf(x1, tv); s1 += x1;
    a2 += fminf(x2, tv); s2 += x2;
    a3 += fminf(x3, tv); s3 += x3;
  }
  relx[(size_t)(n0 + nl + 0) * M_DIM + m0 + ml] = a0 / s0;
  relx[(size_t)(n0 + nl + 1) * M_DIM + m0 + ml] = a1 / s1;
  relx[(size_t)(n0 + nl + 2) * M_DIM + m0 + ml] = a2 / s2;
  relx[(size_t)(n0 + nl + 3) * M_DIM + m0 + ml] = a3 / s3;
}

// ---- Kernel C (fused): rel_w -> ind_x -> chosen_x --------------------------
// rel_w[b,m] = sum_n min(w[n,m], t[b,m]) / sum_n w[n,m]
// ind_x[b,m] = argmax_n min(rel_w[b,m], w[n,m])   (first max, strict >)
// chosen_x[b,m] = min(x[b,ind], w[ind,m])
// grid (M/32, B/32) = 128 blocks for full-chip occupancy, block 256.
// LDS: w column tile 512x32 = 64 KB, scanned: 1x for sw, 2x per b.
__global__ __launch_bounds__(256) void relw_indx_kernel(const float* __restrict__ x,
                                                        const float* __restrict__ w,
                                                        const float* __restrict__ t,
                                                        float* __restrict__ chosen_x) {
  __shared__ float wc[N_DIM * 32];   // [n][m_local]
  const int m0 = blockIdx.x * 32;
  const int b0 = blockIdx.y * 32;
  load_tile(wc, w + m0, M_DIM, N_DIM, 32, N_DIM, M_DIM);
  tile_fence();

  const int ml = threadIdx.x & 31;
  const int bg = threadIdx.x >> 5;             // 8 groups of 4 b's
  const int m = m0 + ml;
  // Denominator sum_n w[n,m]: one LDS column scan (amortized over 4 b's).
  float swm = 0.f;
  for (int n = 0; n < N_DIM; ++n) swm += wc[n * 32 + ml];

  for (int bj = 0; bj < 4; ++bj) {
    const int b = b0 + bg * 4 + bj;
    const float tb = t[(size_t)b * M_DIM + m];
    float s = 0.f;
    for (int n = 0; n < N_DIM; ++n) s += fminf(wc[n * 32 + ml], tb);
    const float relw = s / swm;
    float best = -3.402823466e38f;
    int bi = 0;
    for (int n = 0; n < N_DIM; ++n) {
      const float v = fminf(relw, wc[n * 32 + ml]);
      if (v > best) { best = v; bi = n; }
    }
    chosen_x[(size_t)b * M_DIM + m] = fminf(x[(size_t)b * N_DIM + bi],
                                            wc[bi * 32 + ml]);
  }
}

// ---- Kernel D: ind_w -> chosen_w -------------------------------------------
// ind_w[b,m] = argmax_n min(x[b,n], rel_x[n,m]);  chosen_w = min(x[b,ind], w[ind,m])
// grid (M/32, B/32) = 128 blocks, block 256. LDS: 32 full x rows = 64 KB.
__global__ __launch_bounds__(256) void indw_kernel(const float* __restrict__ x,
                                                   const float* __restrict__ w,
                                                   const float* __restrict__ relx,
                                                   float* __restrict__ chosen_w) {
  __shared__ float xr[32 * N_DIM];   // [b_local][n]
  const int m0 = blockIdx.x * 32;
  const int b0 = blockIdx.y * 32;
  load_tile(xr, x + (size_t)b0 * N_DIM, N_DIM, B_DIM, N_DIM, 32, N_DIM);
  tile_fence();

  const int ml = threadIdx.x & 31;
  const int bl = (threadIdx.x >> 5) * 4;       // 4 b rows per thread
  const int m = m0 + ml;
  float best[4];
  int bi[4];
#pragma unroll
  for (int j = 0; j < 4; ++j) { best[j] = -3.402823466e38f; bi[j] = 0; }
  for (int n = 0; n < N_DIM; ++n) {
    const float rv = relx[(size_t)n * M_DIM + m];   // coalesced, L0/L2-hot
#pragma unroll
    for (int j = 0; j < 4; ++j) {
      const float v = fminf(xr[(bl + j) * N_DIM + n], rv);  // LDS broadcast
      if (v > best[j]) { best[j] = v; bi[j] = n; }
    }
  }
#pragma unroll
  for (int j = 0; j < 4; ++j) {
    const int b = b0 + bl + j;
    chosen_w[(size_t)b * M_DIM + m] =
        fminf(xr[(bl + j) * N_DIM + bi[j]], w[(size_t)bi[j] * M_DIM + m]);
  }
}

extern "C" void kernel_launch(void* const* d_in, const int* in_sizes, int n_in,
                              void* d_out, int out_size, void* d_ws, size_t ws_size,
                              hipStream_t stream) {
  (void)in_sizes; (void)n_in; (void)out_size; (void)ws_size;
  const float* x = (const float*)d_in[0];   // (B, N)
  const float* w = (const float*)d_in[1];   // (N, M)
  const float* t = (const float*)d_in[2];   // (B, M)
  float* out = (float*)d_out;               // chosen_x (B,M) then chosen_w (B,M)
  float* relx = (float*)d_ws;               // N*M floats (1 MB, L2-resident)

  relx_kernel<<<dim3(M_DIM / 32, N_DIM / 32), 256, 0, stream>>>(x, t, relx);
  relw_indx_kernel<<<dim3(M_DIM / 32, B_DIM / 32), 256, 0, stream>>>(x, w, t, out);
  indw_kernel<<<dim3(M_DIM / 32, B_DIM / 32), 256, 0, stream>>>(
      x, w, relx, out + (size_t)B_DIM * M_DIM);
}